// EventPillarsScatter_88278757802226
// MI455X (gfx1250) — compile-verified
//
#include <hip/hip_runtime.h>
#include <hip/hip_bf16.h>
#include <stdint.h>

// Problem constants (from the reference): canvas 512x512, C=64 channels.
#define NYX   (512 * 512)   // 262144 flat BEV cells
#define NCH   64
#define TILE  256           // flat cells per block in the gather kernel

typedef float v4f __attribute__((ext_vector_type(4)));
typedef int   v4i __attribute__((ext_vector_type(4)));

// ---- gfx1250 async global->LDS path selection -------------------------------
#if defined(__gfx1250__)
  #if __has_builtin(__builtin_amdgcn_global_load_async_to_lds_b32)
    #define USE_ASYNC_BUILTIN 1
  #else
    #define USE_ASYNC_ASM 1
  #endif
#endif

__device__ __forceinline__ void wait_asynccnt0() {
#if defined(__gfx1250__)
  #if __has_builtin(__builtin_amdgcn_s_wait_asynccnt)
    __builtin_amdgcn_s_wait_asynccnt(0);
  #else
    asm volatile("s_wait_asynccnt 0" ::: "memory");
  #endif
#endif
}

// ---- phase 1: inverse map init (inv[] = -1), 1 int4 per thread --------------
__global__ void __launch_bounds__(256)
inv_init_kernel(int* __restrict__ inv) {
  const int i = blockIdx.x * blockDim.x + threadIdx.x;   // 65536 threads
  v4i m; m.x = -1; m.y = -1; m.z = -1; m.w = -1;
  ((v4i*)inv)[i] = m;
}

// ---- phase 2: scatter pillar ids into the inverse map -----------------------
__global__ void __launch_bounds__(256)
inv_scatter_kernel(const int* __restrict__ coords, int* __restrict__ inv, int n) {
  const int i = blockIdx.x * blockDim.x + threadIdx.x;
  if (i < n) {
    const int y = coords[3 * i + 1];
    const int x = coords[3 * i + 2];
    inv[(y << 9) + x] = i;
  }
}

// ---- phase 3: gather-formulated scatter; writes every output byte once ------
// Block owns TILE consecutive flat cells for all 64 channels.
// inv tile is staged into LDS with the gfx1250 async copy engine.
__global__ void __launch_bounds__(TILE)
scatter_gather_kernel(const float* __restrict__ feat,
                      const int*   __restrict__ inv,
                      float*       __restrict__ out) {
  __shared__ int tile[TILE];
  const int tid       = threadIdx.x;
  const int flat_base = blockIdx.x * TILE;
  const int* gsrc     = inv + flat_base + tid;

#if defined(USE_ASYNC_BUILTIN)
  __builtin_amdgcn_global_load_async_to_lds_b32(
      (__attribute__((address_space(1))) int*)gsrc,
      (__attribute__((address_space(3))) int*)&tile[tid],
      0, 0);
  wait_asynccnt0();
#elif defined(USE_ASYNC_ASM)
  {
    unsigned lds_off =
        (unsigned)(uintptr_t)(__attribute__((address_space(3))) int*)&tile[tid];
    asm volatile("global_load_async_to_lds_b32 %0, %1, off"
                 :: "v"(lds_off), "v"(gsrc) : "memory");
    wait_asynccnt0();
  }
#else
  tile[tid] = *gsrc;
#endif
  __syncthreads();

  const int   n    = tile[tid];
  const int   flat = flat_base + tid;
  float*      ob   = out + flat;
  const v4f*  frow = (const v4f*)(feat + (int64_t)n * NCH);

#pragma unroll
  for (int cg = 0; cg < NCH / 4; ++cg) {
    v4f v = {0.0f, 0.0f, 0.0f, 0.0f};
    if (n >= 0) v = frow[cg];               // 16B gather from L2-resident feat
    float* o = ob + (size_t)(4 * cg) * NYX; // 4 channel planes, coalesced/wave
    __builtin_nontemporal_store(v.x, o);
    __builtin_nontemporal_store(v.y, o + (size_t)NYX);
    __builtin_nontemporal_store(v.z, o + (size_t)2 * NYX);
    __builtin_nontemporal_store(v.w, o + (size_t)3 * NYX);
  }
}

// ---- fallback path (if workspace is too small for the 1MB inverse map) ------
__global__ void __launch_bounds__(256)
zero_canvas_kernel(float* __restrict__ out) {
  const int i = blockIdx.x * blockDim.x + threadIdx.x;
  v4f z = {0.0f, 0.0f, 0.0f, 0.0f};
  __builtin_nontemporal_store(z, ((v4f*)out) + i);
}

__global__ void __launch_bounds__(256)
direct_scatter_kernel(const float* __restrict__ feat,
                      const int*   __restrict__ coords,
                      float*       __restrict__ out, int n) {
  const int t  = blockIdx.x * blockDim.x + threadIdx.x;
  const int i  = t >> 4;   // pillar
  const int cg = t & 15;   // channel group of 4
  if (i < n) {
    const int flat = (coords[3 * i + 1] << 9) + coords[3 * i + 2];
    const v4f v = ((const v4f*)(feat + (int64_t)i * NCH))[cg];
    float* o = out + (size_t)(4 * cg) * NYX + flat;
    __builtin_nontemporal_store(v.x, o);
    __builtin_nontemporal_store(v.y, o + (size_t)NYX);
    __builtin_nontemporal_store(v.z, o + (size_t)2 * NYX);
    __builtin_nontemporal_store(v.w, o + (size_t)3 * NYX);
  }
}

extern "C" void kernel_launch(void* const* d_in, const int* in_sizes, int n_in,
                              void* d_out, int out_size, void* d_ws, size_t ws_size,
                              hipStream_t stream) {
  const float* feat   = (const float*)d_in[0];  // [N, 64] f32
  const int*   coords = (const int*)d_in[1];    // [N, 3]  i32
  float*       out    = (float*)d_out;          // [64, 512*512] f32
  const int    n      = in_sizes[1] / 3;

  if (ws_size >= (size_t)NYX * sizeof(int)) {
    int* inv = (int*)d_ws;
    inv_init_kernel<<<NYX / 4 / 256, 256, 0, stream>>>(inv);
    inv_scatter_kernel<<<(n + 255) / 256, 256, 0, stream>>>(coords, inv, n);
    scatter_gather_kernel<<<NYX / TILE, TILE, 0, stream>>>(feat, inv, out);
  } else {
    zero_canvas_kernel<<<(NCH * NYX / 4) / 256, 256, 0, stream>>>(out);
    direct_scatter_kernel<<<(n * 16 + 255) / 256, 256, 0, stream>>>(feat, coords, out, n);
  }
}